// HGTLayer_53601191854771
// MI455X (gfx1250) — compile-verified
//
#include <hip/hip_runtime.h>
#include <math.h>

// Problem constants (match reference setup_inputs)
#define NN 16384   // nodes
#define NE 8192    // hyperedges
#define MD 32      // max degree (slots per edge)
#define CI 128     // in_dim == out_ch
#define ED 64      // edge feature dim
#define NH 8       // heads
#define DH 16      // dim per head
#define FF 256     // FFN hidden

typedef float v2f __attribute__((ext_vector_type(2)));
typedef float v8f __attribute__((ext_vector_type(8)));

__device__ __forceinline__ void atomicMaxF(float* addr, float val) {
  // Correct float atomic-max given init = -inf (no NaN inputs expected).
  if (val >= 0.0f) atomicMax((int*)addr, __float_as_int(val));
  else             atomicMin((unsigned int*)addr, __float_as_uint(val));
}

// ---------------------------------------------------------------- fill
__global__ void fill_f32(float* __restrict__ p, float v, int n) {
  int i = blockIdx.x * blockDim.x + threadIdx.x;
  if (i < n) p[i] = v;
}

// ------------------------------------------------- edge list build from H
// H is [NN, NE] row-major; column e has <= MD ones. Slot order within an
// edge is irrelevant (all downstream ops are order-invariant reductions).
__global__ void build_edges(const float4* __restrict__ H4,
                            int* __restrict__ deg, int* __restrict__ idx) {
  int i = blockIdx.x * blockDim.x + threadIdx.x;       // over NN*NE/4
  float4 h = H4[i];
  int base = i << 2;
  int n  = base >> 13;          // / NE
  int e0 = base & (NE - 1);
  float vv[4] = {h.x, h.y, h.z, h.w};
#pragma unroll
  for (int j = 0; j < 4; ++j) {
    if (vv[j] != 0.0f) {
      int s = atomicAdd(&deg[e0 + j], 1);
      idx[(e0 + j) * MD + s] = n;
    }
  }
}

// ------------------------------------------------- WMMA f32 GEMM (generic)
// C[M,N] = A[M,K] @ B[K,N] (+bias[N]) (relu?) (+res[M,N])
// One wave per 16x16 tile, K-loop of V_WMMA_F32_16X16X4_F32.
__global__ void wmma_gemm(const float* __restrict__ A, const float* __restrict__ B,
                          const float* __restrict__ bias, const float* __restrict__ res,
                          float* __restrict__ C, int M, int N, int K, int relu) {
  int wave  = (blockIdx.x * blockDim.x + threadIdx.x) >> 5;
  int lane  = threadIdx.x & 31;
  int ntile = N >> 4;
  int mt = wave / ntile;
  int nt = wave - mt * ntile;
  int row   = (mt << 4) + (lane & 15);   // A row this lane supplies
  int col   = (nt << 4) + (lane & 15);   // B/C column this lane supplies
  int khalf = (lane >> 4) << 1;          // lanes 16-31 hold K+2,K+3

  v8f acc = {};
#pragma unroll 4
  for (int k0 = 0; k0 < K; k0 += 4) {
    int ka = k0 + khalf;
    v2f a, b;
    a.x = A[row * K + ka];
    a.y = A[row * K + ka + 1];
    b.x = B[ka * N + col];
    b.y = B[(ka + 1) * N + col];
    acc = __builtin_amdgcn_wmma_f32_16x16x4_f32(
        /*neg_a=*/false, a, /*neg_b=*/false, b,
        /*c_mod=*/(short)0, acc, /*reuse_a=*/false, /*reuse_b=*/false);
  }

  // C/D layout: VGPR r, lane l -> M = r + 8*(l/16), N = l%16
  int mbase = (mt << 4) + ((lane >> 4) << 3);
  float bv = bias ? bias[col] : 0.0f;
#pragma unroll
  for (int r = 0; r < 8; ++r) {
    int m = mbase + r;
    float vv = acc[r] + bv;
    if (relu) vv = vv > 0.0f ? vv : 0.0f;
    if (res)  vv += res[m * N + col];
    C[m * N + col] = vv;
  }
}

// ------------------------------------------------- hyperedge attention bias
// he_feat[e] = sum_{members} edge_feat[n]; edge_attn = he_feat @ We + be
__global__ void edge_attn_k(const int* __restrict__ idx, const int* __restrict__ deg,
                            const float* __restrict__ ef, const float* __restrict__ We,
                            const float* __restrict__ be, float* __restrict__ eattn) {
  __shared__ float he[ED];
  int e = blockIdx.x, t = threadIdx.x;   // 64 threads
  int dg = deg[e];
  float acc = 0.0f;
  for (int s = 0; s < dg; ++s) acc += ef[idx[e * MD + s] * ED + t];
  he[t] = acc;
  __syncthreads();
  if (t < NH) {
    float a = be[t];
#pragma unroll 8
    for (int j = 0; j < ED; ++j) a += he[j] * We[j * NH + t];
    eattn[e * NH + t] = a;
  }
}

// ------------------------------------------------- scores + segment max
__global__ void score_k(const int* __restrict__ idx, const int* __restrict__ deg,
                        const float* __restrict__ qd, const float* __restrict__ kd,
                        const float* __restrict__ eattn,
                        float* __restrict__ sc, float* __restrict__ seg_max) {
  int e = blockIdx.x, t = threadIdx.x;   // 256 = MD*NH
  int s = t >> 3, h = t & 7;
  if (s >= deg[e]) return;
  int node = idx[e * MD + s];
  const float* qr = qd + node * CI + h * DH;
  const float* kr = kd + node * CI + h * DH;
  float d = 0.0f;
#pragma unroll
  for (int j = 0; j < DH; ++j) d += qr[j] * kr[j];
  float v = d * 0.25f + eattn[e * NH + h];   // scale = 1/sqrt(16)
  sc[(e * MD + s) * NH + h] = v;
  atomicMaxF(&seg_max[node * NH + h], v);
}

// ------------------------------------------------- exp + segment sum
__global__ void exp_k(const int* __restrict__ idx, const int* __restrict__ deg,
                      float* __restrict__ sc, const float* __restrict__ seg_max,
                      float* __restrict__ seg_sum) {
  int e = blockIdx.x, t = threadIdx.x;
  int s = t >> 3, h = t & 7;
  if (s >= deg[e]) return;
  int node = idx[e * MD + s];
  float ex = expf(sc[(e * MD + s) * NH + h] - seg_max[node * NH + h]);
  sc[(e * MD + s) * NH + h] = ex;
  atomicAdd(&seg_sum[node * NH + h], ex);
}

// ------------------------------------------------- per-edge message sums
__global__ void msg_k(const int* __restrict__ idx, const int* __restrict__ deg,
                      const float* __restrict__ sc, const float* __restrict__ seg_sum,
                      const float* __restrict__ vd, float* __restrict__ msg) {
  int e = blockIdx.x, c = threadIdx.x;   // 128 threads
  int h = c >> 4;
  int dg = deg[e];
  float acc = 0.0f;
  for (int s = 0; s < dg; ++s) {
    int node = idx[e * MD + s];
    float a = sc[(e * MD + s) * NH + h] / (seg_sum[node * NH + h] + 1e-10f);
    acc += a * vd[node * CI + c];
  }
  msg[e * CI + c] = acc;
}

// ------------------------------------------------- rst = H @ msg (scatter)
__global__ void scatter_k(const int* __restrict__ idx, const int* __restrict__ deg,
                          const float* __restrict__ msg, float* __restrict__ rst) {
  long long tid = (long long)blockIdx.x * 256 + threadIdx.x;  // E*MD*CI threads
  int c = (int)(tid & (CI - 1));
  int s = (int)((tid >> 7) & (MD - 1));
  int e = (int)(tid >> 12);
  if (s >= deg[e]) return;
  atomicAdd(&rst[idx[e * MD + s] * CI + c], msg[e * CI + c]);
}

// ----------------------------------------------------------------- launch
extern "C" void kernel_launch(void* const* d_in, const int* in_sizes, int n_in,
                              void* d_out, int out_size, void* d_ws, size_t ws_size,
                              hipStream_t stream) {
  const float* q  = (const float*)d_in[0];
  const float* k  = (const float*)d_in[1];
  const float* v  = (const float*)d_in[2];
  const float* ef = (const float*)d_in[3];
  const float* H  = (const float*)d_in[4];
  const float* Wq = (const float*)d_in[5];
  const float* Wk = (const float*)d_in[6];
  const float* Wv = (const float*)d_in[7];
  const float* We = (const float*)d_in[8];
  const float* be = (const float*)d_in[9];
  const float* Wo = (const float*)d_in[10];
  const float* W1 = (const float*)d_in[11];
  const float* b1 = (const float*)d_in[12];
  const float* W2 = (const float*)d_in[13];
  const float* b2 = (const float*)d_in[14];
  float* out = (float*)d_out;

  // -------- workspace carve-out (all 256B aligned)
  size_t off = 0;
  auto carve = [&](size_t bytes) -> void* {
    void* p = (char*)d_ws + off;
    off += (bytes + 255) & ~(size_t)255;
    return p;
  };
  float* qd      = (float*)carve((size_t)NN * CI * 4);
  float* kd      = (float*)carve((size_t)NN * CI * 4);
  float* vd      = (float*)carve((size_t)NN * CI * 4);
  float* rst_agg = (float*)carve((size_t)NN * CI * 4);
  float* rst2    = (float*)carve((size_t)NN * CI * 4);
  float* h1      = (float*)carve((size_t)NN * FF * 4);
  float* msg     = (float*)carve((size_t)NE * CI * 4);
  float* sc      = (float*)carve((size_t)NE * MD * NH * 4);
  float* eattn   = (float*)carve((size_t)NE * NH * 4);
  float* seg_max = (float*)carve((size_t)NN * NH * 4);
  float* seg_sum = (float*)carve((size_t)NN * NH * 4);
  int*   idx     = (int*)  carve((size_t)NE * MD * 4);
  int*   deg     = (int*)  carve((size_t)NE * 4);
  (void)ws_size; (void)in_sizes; (void)n_in; (void)out_size;

  // -------- init
  hipMemsetAsync(deg, 0, (size_t)NE * 4, stream);
  hipMemsetAsync(seg_sum, 0, (size_t)NN * NH * 4, stream);
  hipMemsetAsync(rst_agg, 0, (size_t)NN * CI * 4, stream);
  fill_f32<<<(NN * NH + 255) / 256, 256, 0, stream>>>(seg_max, -INFINITY, NN * NH);

  // -------- edge membership from dense H (512 MB scan, HBM-bound)
  {
    int quads = NN * NE / 4;
    build_edges<<<quads / 256, 256, 0, stream>>>((const float4*)H, deg, idx);
  }

  // -------- projections qd/kd/vd = x @ W (WMMA f32)
  {
    int blocks = (NN / 16) * (CI / 16) / 4;   // 4 waves/block
    wmma_gemm<<<blocks, 128, 0, stream>>>(q, Wq, nullptr, nullptr, qd, NN, CI, CI, 0);
    wmma_gemm<<<blocks, 128, 0, stream>>>(k, Wk, nullptr, nullptr, kd, NN, CI, CI, 0);
    wmma_gemm<<<blocks, 128, 0, stream>>>(v, Wv, nullptr, nullptr, vd, NN, CI, CI, 0);
  }

  // -------- hyperedge attention bias
  edge_attn_k<<<NE, ED, 0, stream>>>(idx, deg, ef, We, be, eattn);

  // -------- scores + segment softmax over nodes
  score_k<<<NE, MD * NH, 0, stream>>>(idx, deg, qd, kd, eattn, sc, seg_max);
  exp_k<<<NE, MD * NH, 0, stream>>>(idx, deg, sc, seg_max, seg_sum);

  // -------- messages per edge, then rst = H @ msg via atomic scatter
  msg_k<<<NE, CI, 0, stream>>>(idx, deg, sc, seg_sum, vd, msg);
  scatter_k<<<(NE * MD * CI) / 256, 256, 0, stream>>>(idx, deg, msg, rst_agg);

  // -------- output projection + residual, FFN, second residual (WMMA f32)
  {
    int blocks128 = (NN / 16) * (CI / 16) / 4;
    int blocks256 = (NN / 16) * (FF / 16) / 4;
    // rst2 = rst_agg @ Wo + q
    wmma_gemm<<<blocks128, 128, 0, stream>>>(rst_agg, Wo, nullptr, q, rst2, NN, CI, CI, 0);
    // h1 = relu(rst2 @ W1 + b1)
    wmma_gemm<<<blocks256, 128, 0, stream>>>(rst2, W1, b1, nullptr, h1, NN, FF, CI, 1);
    // out = h1 @ W2 + b2 + rst2
    wmma_gemm<<<blocks128, 128, 0, stream>>>(h1, W2, b2, rst2, out, NN, CI, FF, 0);
  }
}